// MyViT_20753281974992
// MI455X (gfx1250) — compile-verified
//
#include <hip/hip_runtime.h>
#include <hip/hip_bf16.h>
#include <stdint.h>

// Reference = depatchify(patchify(x)) which is an exact inverse pair -> identity.
// Optimal kernel is a pure HBM-rate copy: 512 MB total traffic, ~22 us at 23.3 TB/s.
// CDNA5 path: async global<->LDS DMA (ASYNCcnt) with 128-bit transfers.

typedef __attribute__((ext_vector_type(4))) float f4;
typedef int v4i __attribute__((vector_size(4 * sizeof(int))));  // matches builtin param type

#define AS1 __attribute__((address_space(1)))
#define AS3 __attribute__((address_space(3)))

#if defined(__AMDGCN__) && \
    __has_builtin(__builtin_amdgcn_global_load_async_to_lds_b128) && \
    __has_builtin(__builtin_amdgcn_global_store_async_from_lds_b128)
#define USE_ASYNC_LDS 1
#else
#define USE_ASYNC_LDS 0
#endif

__device__ __forceinline__ void wait_async_0() {
#if __has_builtin(__builtin_amdgcn_s_wait_asynccnt)
    __builtin_amdgcn_s_wait_asynccnt(0);
#else
    asm volatile("s_wait_asynccnt 0" ::: "memory");
#endif
}

__global__ __launch_bounds__(256) void vit_identity_copy(const f4* __restrict__ in,
                                                         f4* __restrict__ out,
                                                         unsigned long long n16) {
    const unsigned long long stride = (unsigned long long)gridDim.x * blockDim.x;
    const unsigned long long i0 = (unsigned long long)blockIdx.x * blockDim.x + threadIdx.x;

#if USE_ASYNC_LDS
    // Each thread owns a private 16B slot in each of two LDS tile buffers.
    // No cross-thread LDS sharing -> per-wave ASYNCcnt ordering suffices, no barriers.
    __shared__ char ldsbuf[2 * 256 * 16];
    AS3 char* lds = (AS3 char*)ldsbuf;
    AS3 v4i* l0 = (AS3 v4i*)(lds + (unsigned)threadIdx.x * 16u);
    AS3 v4i* l1 = (AS3 v4i*)(lds + 256 * 16 + (unsigned)threadIdx.x * 16u);

    for (unsigned long long idx = i0; idx < n16; idx += 2ull * stride) {
        const unsigned long long idx2 = idx + stride;
        // Stage two 16B/lane async loads into LDS (ASYNCcnt += 2 per wave).
        __builtin_amdgcn_global_load_async_to_lds_b128(
            (AS1 v4i*)(in + idx), l0, 0, 0);
        if (idx2 < n16) {
            __builtin_amdgcn_global_load_async_to_lds_b128(
                (AS1 v4i*)(in + idx2), l1, 0, 0);
        }
        wait_async_0();  // loads landed in LDS
        // Stream them back out LDS -> global, still no VGPR data path.
        __builtin_amdgcn_global_store_async_from_lds_b128(
            (AS1 v4i*)(out + idx), l0, 0, 0);
        if (idx2 < n16) {
            __builtin_amdgcn_global_store_async_from_lds_b128(
                (AS1 v4i*)(out + idx2), l1, 0, 0);
        }
        wait_async_0();  // stores drained before LDS slots are reused
    }
#else
    // Fallback: plain 128-bit nontemporal streaming copy (also HBM-limited).
    for (unsigned long long idx = i0; idx < n16; idx += stride) {
        f4 v = __builtin_nontemporal_load(in + idx);
        __builtin_nontemporal_store(v, out + idx);
    }
#endif
}

// Scalar tail (not needed for 64*1*1024*1024, but keeps the kernel general).
__global__ void vit_identity_tail(const float* __restrict__ in, float* __restrict__ out,
                                  unsigned long long start, unsigned long long n) {
    unsigned long long i = start + (unsigned long long)blockIdx.x * blockDim.x + threadIdx.x;
    if (i < n) out[i] = in[i];
}

extern "C" void kernel_launch(void* const* d_in, const int* in_sizes, int n_in,
                              void* d_out, int out_size, void* d_ws, size_t ws_size,
                              hipStream_t stream) {
    (void)d_ws; (void)ws_size; (void)n_in; (void)in_sizes;

    const float* in = (const float*)d_in[0];   // images, fp32
    float* out = (float*)d_out;                // identity of images
    // d_in[1] (n_patches) is irrelevant: patchify/depatchify compose to identity.

    const unsigned long long n   = (unsigned long long)out_size;  // 64 Mi floats
    const unsigned long long n16 = n / 4;                          // 16 Mi float4
    const unsigned long long rem = n - n16 * 4;

    if (n16) {
        const int threads = 256;
        const unsigned long long per_block = (unsigned long long)threads * 8ull; // 8 x b128 per thread
        unsigned long long blocks = (n16 + per_block - 1) / per_block;
        if (blocks == 0) blocks = 1;
        vit_identity_copy<<<(unsigned)blocks, threads, 0, stream>>>(
            (const f4*)in, (f4*)out, n16);
    }
    if (rem) {
        vit_identity_tail<<<1, 256, 0, stream>>>(in, out, n16 * 4, n);
    }
}